// GNNNode_52931176955957
// MI455X (gfx1250) — compile-verified
//
#include <hip/hip_runtime.h>

// Problem constants (match reference)
#define N_NODES 50000
#define N_EDGES 800000
#define DIM     64
#define LAYERS  5
#define NFEAT   9
#define EFEAT   3
#define AVOC    128
#define BVOC    8
#define BN_EPS  1e-5f

typedef __attribute__((ext_vector_type(2))) float v2f;
typedef __attribute__((ext_vector_type(8))) float v8f;

// ---------------------------------------------------------------------------
// Atom encoder: h[n,d] = sum_f atom_emb[f, x[n,f], d]
// ---------------------------------------------------------------------------
__global__ void atom_embed_kernel(const int* __restrict__ x,
                                  const float* __restrict__ atom_emb,
                                  float* __restrict__ h) {
    int tid = blockIdx.x * blockDim.x + threadIdx.x;
    if (tid >= N_NODES * DIM) return;
    int n = tid >> 6;
    int d = tid & 63;
    float acc = 0.0f;
#pragma unroll
    for (int f = 0; f < NFEAT; ++f) {
        int v = x[n * NFEAT + f];
        acc += atom_emb[(f * AVOC + v) * DIM + d];
    }
    h[tid] = acc;
}

// ---------------------------------------------------------------------------
// Edge messages: agg[dst] += relu(h[src] + bond_embed(edge_attr))
// 16 threads per edge, each handles a float4 of the 64-dim channel.
// Bond tables (3*8*64 floats = 6 KB) staged in LDS via CDNA5 async
// Global->LDS copy (ASYNCcnt path), completed before edge-index loads.
// ---------------------------------------------------------------------------
__global__ void edge_msg_kernel(const int* __restrict__ edge_index,
                                const int* __restrict__ edge_attr,
                                const float* __restrict__ bond_emb_l,
                                const float* __restrict__ h,
                                float* __restrict__ agg) {
    __shared__ float bsm[EFEAT * BVOC * DIM];   // 6144 bytes

    // Async copy 6144 B in 16-byte chunks (384 chunks, 256 threads).
    {
        const uint32_t lds_base = (uint32_t)(uintptr_t)&bsm[0];
        for (int ck = threadIdx.x; ck < (EFEAT * BVOC * DIM * 4) / 16; ck += blockDim.x) {
            uint32_t loff = lds_base + ck * 16;   // LDS byte address (VDST)
            uint32_t goff = ck * 16;              // byte offset from SGPR base
            asm volatile("global_load_async_to_lds_b128 %0, %1, %2 offset:0"
                         :: "v"(loff), "v"(goff), "s"(bond_emb_l)
                         : "memory");
        }
        asm volatile("s_wait_asynccnt 0x0" ::: "memory");
    }
    __syncthreads();

    int tid = blockIdx.x * blockDim.x + threadIdx.x;
    if (tid >= N_EDGES * 16) return;
    int e = tid >> 4;
    int q = (tid & 15) * 4;

    int src = edge_index[e];
    int dst = edge_index[N_EDGES + e];
    int a0 = edge_attr[e * EFEAT + 0];
    int a1 = edge_attr[e * EFEAT + 1];
    int a2 = edge_attr[e * EFEAT + 2];

    const float4 hv = *(const float4*)(h + src * DIM + q);
    float m[4] = {hv.x, hv.y, hv.z, hv.w};
#pragma unroll
    for (int j = 0; j < 4; ++j) {
        float eb = bsm[(0 * BVOC + a0) * DIM + q + j]
                 + bsm[(1 * BVOC + a1) * DIM + q + j]
                 + bsm[(2 * BVOC + a2) * DIM + q + j];
        float msg = fmaxf(m[j] + eb, 0.0f);
        atomicAdd(agg + dst * DIM + q + j, msg);
    }
}

// ---------------------------------------------------------------------------
// Fused GEMM: out[N,64] = stage(in) @ W + bias, plus BN partial sums.
//   MODE 0: stage = (1 + eps[layer]) * in0 + in1          (GIN combine -> MLP1)
//   MODE 1: stage = relu(bn1(in0)) using stats_in          (MLP2)
// Block = 256 threads (8 waves). Block tile = 32 rows x 64 cols; each wave
// computes one 16x16 tile via 16x V_WMMA_F32_16X16X4_F32 (K=64, fp32 exact).
// LDS rows padded to 66 floats to spread banks; W stored transposed so both
// A and B operands are contiguous float2 reads matching the ISA VGPR layout:
//   A (16x4 f32): lanes 0-15 -> {K0,K1}, lanes 16-31 -> {K2,K3}
//   B (4x16 f32): lanes 0-15 -> {K0,K1}@col, lanes 16-31 -> {K2,K3}@col
// ---------------------------------------------------------------------------
#define LDSW 66

template <int MODE>
__global__ void gemm_kernel(const float* __restrict__ in0,
                            const float* __restrict__ in1,        // agg (MODE 0)
                            const float* __restrict__ eps_or_stats,
                            const float* __restrict__ gamma,      // MODE 1
                            const float* __restrict__ beta,       // MODE 1
                            const float* __restrict__ W,          // [64][64] row-major
                            const float* __restrict__ bias,       // [64]
                            float* __restrict__ out,
                            float* __restrict__ stats_out,        // [2][64]
                            int layer) {
    __shared__ float zs[32 * LDSW];   // staged input tile
    __shared__ float wt[64 * LDSW];   // W transposed: wt[n][k] = W[k][n]

    const int t  = threadIdx.x;
    const int m0 = blockIdx.x * 32;
    const int c  = t & 63;            // column handled by this thread (staging)
    const int r0 = t >> 6;            // 0..3

    // Load W transposed into LDS (coalesced global reads, n = c fixed).
#pragma unroll
    for (int i = 0; i < 16; ++i) {
        int k = r0 + i * 4;
        wt[c * LDSW + k] = W[k * DIM + c];
    }

    // Hoist per-column staging parameters (c is loop-invariant).
    float sc = 0.0f, mu = 0.0f, rstd = 0.0f, gm = 0.0f, bt = 0.0f;
    if (MODE == 0) {
        sc = 1.0f + eps_or_stats[layer];
    } else {
        const float inv_n = 1.0f / (float)N_NODES;
        mu = eps_or_stats[c] * inv_n;
        float var = eps_or_stats[DIM + c] * inv_n - mu * mu;
        rstd = rsqrtf(var + BN_EPS);
        gm = gamma[c];
        bt = beta[c];
    }

    // Stage 32x64 input tile with the fused pre-op.
#pragma unroll
    for (int i = 0; i < 8; ++i) {
        int r = r0 + i * 4;
        int g = m0 + r;
        float v = 0.0f;
        if (g < N_NODES) {
            float x0 = in0[g * DIM + c];
            if (MODE == 0) {
                v = sc * x0 + in1[g * DIM + c];
            } else {
                v = fmaxf(gm * (x0 - mu) * rstd + bt, 0.0f);
            }
        }
        zs[r * LDSW + c] = v;
    }
    __syncthreads();

    const int wv     = t >> 5;        // wave 0..7
    const int lane   = t & 31;
    const int half   = lane >> 4;     // 0: K pair {0,1}; 1: K pair {2,3}
    const int lane16 = lane & 15;
    const int rtile  = wv >> 2;       // 0..1
    const int ctile  = wv & 3;        // 0..3
    const int arow   = rtile * 16 + lane16;   // A-operand row in block tile
    const int ncol   = ctile * 16 + lane16;   // output column

    v8f acc = {};
#pragma unroll
    for (int k = 0; k < 16; ++k) {
        v2f a = *(const v2f*)(&zs[arow * LDSW + 4 * k + 2 * half]);
        v2f b = *(const v2f*)(&wt[ncol * LDSW + 4 * k + 2 * half]);
        acc = __builtin_amdgcn_wmma_f32_16x16x4_f32(
            false, a, false, b, (short)0, acc, false, false);
    }

    // Epilogue. C/D layout: VGPR i -> row (rtile*16 + half*8 + i).
    // N_NODES % 8 == 0, so validity is uniform per half-wave: one exec mask,
    // no per-element branches, and padding tiles skip stores + atomics.
    const int mrow_base = m0 + rtile * 16 + half * 8;
    if (mrow_base < N_NODES) {
        float bv = bias[ncol];
        float s = 0.0f, s2 = 0.0f;
#pragma unroll
        for (int i = 0; i < 8; ++i) {
            float y = acc[i] + bv;
            out[(mrow_base + i) * DIM + ncol] = y;
            s  += y;
            s2 += y * y;
        }
        atomicAdd(&stats_out[ncol], s);
        atomicAdd(&stats_out[DIM + ncol], s2);
    }
}

// ---------------------------------------------------------------------------
// Final BN (+ optional relu) per layer.
// ---------------------------------------------------------------------------
__global__ void bn_final_kernel(const float* __restrict__ z,
                                const float* __restrict__ stats,  // [2][64]
                                const float* __restrict__ gamma,
                                const float* __restrict__ beta,
                                float* __restrict__ out,
                                int do_relu) {
    int tid = blockIdx.x * blockDim.x + threadIdx.x;
    if (tid >= N_NODES * DIM) return;
    int c = tid & 63;
    float inv_n = 1.0f / (float)N_NODES;
    float mu  = stats[c] * inv_n;
    float var = stats[DIM + c] * inv_n - mu * mu;
    float v = gamma[c] * (z[tid] - mu) * rsqrtf(var + BN_EPS) + beta[c];
    out[tid] = do_relu ? fmaxf(v, 0.0f) : v;
}

// ---------------------------------------------------------------------------
// Launch
// ---------------------------------------------------------------------------
extern "C" void kernel_launch(void* const* d_in, const int* in_sizes, int n_in,
                              void* d_out, int out_size, void* d_ws, size_t ws_size,
                              hipStream_t stream) {
    const int*   x          = (const int*)  d_in[0];
    const int*   edge_index = (const int*)  d_in[1];
    const int*   edge_attr  = (const int*)  d_in[2];
    const float* atom_emb   = (const float*)d_in[3];
    const float* bond_emb   = (const float*)d_in[4];
    const float* eps        = (const float*)d_in[5];
    const float* W1         = (const float*)d_in[6];
    const float* b1         = (const float*)d_in[7];
    const float* bn1_gamma  = (const float*)d_in[8];
    const float* bn1_beta   = (const float*)d_in[9];
    const float* W2         = (const float*)d_in[10];
    const float* b2         = (const float*)d_in[11];
    const float* bn_gamma   = (const float*)d_in[12];
    const float* bn_beta    = (const float*)d_in[13];

    float* out = (float*)d_out;

    // Workspace layout (all fp32): h | agg | y | z2 | stats[4*64]
    float* h     = (float*)d_ws;
    float* agg   = h   + (size_t)N_NODES * DIM;
    float* y     = agg + (size_t)N_NODES * DIM;
    float* z2    = y   + (size_t)N_NODES * DIM;
    float* stats = z2  + (size_t)N_NODES * DIM;   // [sum1|sumsq1|sum2|sumsq2]

    const int nd_blocks   = (N_NODES * DIM + 255) / 256;
    const int edge_blocks = (N_EDGES * 16 + 255) / 256;
    const int gemm_blocks = (N_NODES + 31) / 32;

    atom_embed_kernel<<<nd_blocks, 256, 0, stream>>>(x, atom_emb, h);

    for (int l = 0; l < LAYERS; ++l) {
        hipMemsetAsync(agg, 0, (size_t)N_NODES * DIM * sizeof(float), stream);
        hipMemsetAsync(stats, 0, 4 * DIM * sizeof(float), stream);

        edge_msg_kernel<<<edge_blocks, 256, 0, stream>>>(
            edge_index, edge_attr, bond_emb + (size_t)l * EFEAT * BVOC * DIM, h, agg);

        gemm_kernel<0><<<gemm_blocks, 256, 0, stream>>>(
            h, agg, eps, nullptr, nullptr,
            W1 + (size_t)l * DIM * DIM, b1 + (size_t)l * DIM,
            y, stats, l);

        gemm_kernel<1><<<gemm_blocks, 256, 0, stream>>>(
            y, nullptr, stats, bn1_gamma + (size_t)l * DIM, bn1_beta + (size_t)l * DIM,
            W2 + (size_t)l * DIM * DIM, b2 + (size_t)l * DIM,
            z2, stats + 2 * DIM, l);

        float* dst = (l == LAYERS - 1) ? out : h;
        bn_final_kernel<<<nd_blocks, 256, 0, stream>>>(
            z2, stats + 2 * DIM, bn_gamma + (size_t)l * DIM, bn_beta + (size_t)l * DIM,
            dst, (l != LAYERS - 1) ? 1 : 0);
    }
}